// DiffSLIC_71940702208463
// MI455X (gfx1250) — compile-verified
//
#include <hip/hip_runtime.h>

typedef __attribute__((ext_vector_type(2))) float v2f;
typedef __attribute__((ext_vector_type(8))) float v8f;

#define B_    2
#define C_    32
#define H_    256
#define W_    256
#define GS    16          // 16x16 superpixel grid
#define BS    16          // 16x16 pixels per block
#define HW    (H_ * W_)
#define INV_TAU 100.0f

// ------------------------------------------------------------------
// Kernel 1: per-pixel L2 normalize over channels: xn = x / ||x||_c
// ------------------------------------------------------------------
__global__ __launch_bounds__(256) void k_normalize(const float* __restrict__ x,
                                                   float* __restrict__ xn) {
  int idx = blockIdx.x * 256 + threadIdx.x;   // over B*H*W
  int b   = idx >> 16;
  int pix = idx & 0xFFFF;
  const float* px = x + (size_t)b * C_ * HW + pix;
  float v[C_];
  float ss = 0.f;
#pragma unroll
  for (int c = 0; c < C_; ++c) { v[c] = px[c * HW]; ss += v[c] * v[c]; }
  float inv = rsqrtf(ss);
  float* po = xn + (size_t)b * C_ * HW + pix;
#pragma unroll
  for (int c = 0; c < C_; ++c) po[c * HW] = v[c] * inv;
}

// ------------------------------------------------------------------
// Kernel 2: init clusters = block mean of RAW x, then L2-normalized.
// One wave (32 threads == 32 channels) per superpixel.
// ------------------------------------------------------------------
__global__ __launch_bounds__(32) void k_init_clst(const float* __restrict__ x,
                                                  float* __restrict__ clst) {
  int b  = blockIdx.x >> 8;
  int sp = blockIdx.x & 255;
  int i = sp >> 4, j = sp & 15;
  int c = threadIdx.x;
  const float* px = x + ((size_t)(b * C_ + c) * H_ + i * BS) * W_ + j * BS;
  float s = 0.f;
  for (int py = 0; py < BS; ++py)
#pragma unroll
    for (int qx = 0; qx < BS; ++qx) s += px[py * W_ + qx];
  float mean = s * (1.0f / 256.0f);
  float ss = mean * mean;
#pragma unroll
  for (int off = 16; off > 0; off >>= 1) ss += __shfl_xor(ss, off, 32);
  clst[((b * C_ + c) * GS + i) * GS + j] = mean * rsqrtf(ss);
}

// ------------------------------------------------------------------
// Kernel 3: per-block GEMM via V_WMMA_F32_16X16X4_F32:
//   sim[pixel(256) , neighbor-cluster(9, padded to 16)] , K = 32 channels.
// MODE 0: store pix_sim[b][blk][pix][16] for the iteration update.
// MODE 1: fuse 9-way softmax -> write p2s (final assignment).
// One workgroup (256 thr = 8 waves) per (b, block); 2 M-tiles per wave.
// Templated on MODE so each instantiation has branch-free stores.
// ------------------------------------------------------------------
template <int MODE>
__global__ __launch_bounds__(256) void k_block_gemm(const float* __restrict__ xn,
                                                    const float* __restrict__ clst,
                                                    float* __restrict__ out) {
  __shared__ float As[C_][256];                    // [channel][local pixel]
  __shared__ float Bs[C_][16];                     // [channel][neighbor slot]
  __shared__ float simL[MODE ? 256 * 17 : 1];      // only allocated for MODE 1

  int b   = blockIdx.x >> 8;
  int blk = blockIdx.x & 255;
  int i = blk >> 4, j = blk & 15;
  int tid = threadIdx.x;

  // ---- stage A: block's normalized pixels, [c][p]
  {
    int gy = i * BS + (tid >> 4), gx = j * BS + (tid & 15);
    const float* xp = xn + (size_t)b * C_ * HW + gy * W_ + gx;
#pragma unroll
    for (int c = 0; c < C_; ++c) As[c][tid] = xp[c * HW];
  }
  // ---- stage B: 3x3 neighborhood clusters, zero-pad OOB and n=9..15
  for (int idx = tid; idx < C_ * 16; idx += 256) {
    int c = idx >> 4, n = idx & 15;
    float v = 0.f;
    if (n < 9) {
      int ii = i + n / 3 - 1, jj = j + n % 3 - 1;
      if (ii >= 0 && ii < GS && jj >= 0 && jj < GS)
        v = clst[((b * C_ + c) * GS + ii) * GS + jj];
    }
    Bs[c][n] = v;
  }
  __syncthreads();

  int wave = tid >> 5, lane = tid & 31;
  int mrow = lane & 15;
  int koff = (lane >> 4) << 1;   // 0 or 2: K-pair owned by this half-wave

  // B fragments shared by both tiles of this wave
  v2f bfrag[8];
#pragma unroll
  for (int kk = 0; kk < 8; ++kk) {
    int k0 = kk * 4 + koff;
    bfrag[kk].x = Bs[k0][mrow];       // N = lane&15
    bfrag[kk].y = Bs[k0 + 1][mrow];
  }

#pragma unroll
  for (int tt = 0; tt < 2; ++tt) {
    int t = wave * 2 + tt;            // M-tile index 0..15
    v8f acc = {};
#pragma unroll
    for (int kk = 0; kk < 8; ++kk) {
      int k0 = kk * 4 + koff;
      v2f afrag;
      afrag.x = As[k0][t * 16 + mrow];
      afrag.y = As[k0 + 1][t * 16 + mrow];
      acc = __builtin_amdgcn_wmma_f32_16x16x4_f32(
          false, afrag, false, bfrag[kk], (short)0, acc, false, false);
    }
    // D layout: VGPR r, lane l -> row r + 8*(l>=16), col l&15
#pragma unroll
    for (int r = 0; r < 8; ++r) {
      int pix = t * 16 + r + ((lane >> 4) << 3);
      int n   = lane & 15;
      if constexpr (MODE == 0) {
        out[(((size_t)(b * 256 + blk)) * 256 + pix) * 16 + n] = acc[r];
      } else {
        simL[pix * 17 + n] = acc[r];
      }
    }
  }

  if constexpr (MODE == 1) {
    __syncthreads();
    // per-pixel softmax over 9 candidate clusters (OOB -> weight 0)
    int gy = i * BS + (tid >> 4), gx = j * BS + (tid & 15);
    float sv[9];
    bool  va[9];
    float mx = -__builtin_inff();
#pragma unroll
    for (int n = 0; n < 9; ++n) {
      int ii = i + n / 3 - 1, jj = j + n % 3 - 1;
      va[n] = (ii >= 0 && ii < GS && jj >= 0 && jj < GS);
      sv[n] = simL[tid * 17 + n];
      if (va[n]) mx = fmaxf(mx, sv[n]);
    }
    float e[9];
    float sum = 0.f;
#pragma unroll
    for (int n = 0; n < 9; ++n) {
      e[n] = va[n] ? __expf((sv[n] - mx) * INV_TAU) : 0.f;
      sum += e[n];
    }
    float inv = 1.f / sum;
#pragma unroll
    for (int n = 0; n < 9; ++n)
      out[((size_t)b * 9 + n) * HW + gy * W_ + gx] = e[n] * inv;
  }
}

// ------------------------------------------------------------------
// Kernel 4: per-superpixel softmax over its 48x48 window (2304 px),
// write s2p, and weighted+renormalized cluster update.
// One workgroup (256 thr) per (b, superpixel); 9 window px per thread.
// ------------------------------------------------------------------
__global__ __launch_bounds__(256) void k_update(const float* __restrict__ xn,
                                                const float* __restrict__ pix_sim,
                                                float* __restrict__ clst,
                                                float* __restrict__ s2p) {
  __shared__ float red[256];
  __shared__ float accL[256 * 33];   // padded stride to dodge bank conflicts

  int b  = blockIdx.x >> 8;
  int sp = blockIdx.x & 255;
  int i = sp >> 4, j = sp & 15;
  int tid = threadIdx.x;

  float sv[9];
  bool  va[9];
  int   gyA[9], gxA[9];
  float lmax = -__builtin_inff();
#pragma unroll
  for (int q = 0; q < 9; ++q) {
    int p  = tid + 256 * q;               // window index 0..2303
    int wy = p / 48, wx = p % 48;
    int gy = i * BS + wy - 16, gx = j * BS + wx - 16;
    bool v = (gy >= 0 && gy < H_ && gx >= 0 && gx < W_);
    va[q] = v; gyA[q] = gy; gxA[q] = gx;
    float s = -__builtin_inff();
    if (v) {
      int bi = gy >> 4, bj = gx >> 4;
      int lp = ((gy & 15) << 4) | (gx & 15);
      int nidx = (i - bi + 1) * 3 + (j - bj + 1); // slot of cluster (i,j) in that block's GEMM
      s = pix_sim[(((size_t)(b * 256 + bi * 16 + bj)) * 256 + lp) * 16 + nidx];
      lmax = fmaxf(lmax, s);
    }
    sv[q] = s;
  }

  red[tid] = lmax; __syncthreads();
  for (int st = 128; st > 0; st >>= 1) {
    if (tid < st) red[tid] = fmaxf(red[tid], red[tid + st]);
    __syncthreads();
  }
  float gmax = red[0];
  __syncthreads();

  float e[9];
  float lsum = 0.f;
#pragma unroll
  for (int q = 0; q < 9; ++q) {
    e[q] = va[q] ? __expf((sv[q] - gmax) * INV_TAU) : 0.f;
    lsum += e[q];
  }
  red[tid] = lsum; __syncthreads();
  for (int st = 128; st > 0; st >>= 1) {
    if (tid < st) red[tid] += red[tid + st];
    __syncthreads();
  }
  float inv = 1.f / red[0];

  float acc[C_];
#pragma unroll
  for (int c = 0; c < C_; ++c) acc[c] = 0.f;
#pragma unroll
  for (int q = 0; q < 9; ++q) {
    float w = e[q] * inv;                 // 0 for OOB
    int p = tid + 256 * q;
    s2p[((size_t)b * 2304 + p) * 256 + sp] = w;
    if (va[q]) {
      const float* xp = xn + (size_t)b * C_ * HW + gyA[q] * W_ + gxA[q];
#pragma unroll
      for (int c = 0; c < C_; ++c) acc[c] += w * xp[c * HW];
    }
  }

#pragma unroll
  for (int c = 0; c < C_; ++c) accL[tid * 33 + c] = acc[c];
  __syncthreads();
  for (int st = 128; st > 0; st >>= 1) {
    if (tid < st) {
#pragma unroll
      for (int c = 0; c < C_; ++c) accL[tid * 33 + c] += accL[(tid + st) * 33 + c];
    }
    __syncthreads();
  }
  if (tid < 32) {                        // wave 0: channel = tid
    float v = accL[tid];
    float ss = v * v;
#pragma unroll
    for (int off = 16; off > 0; off >>= 1) ss += __shfl_xor(ss, off, 32);
    clst[((b * C_ + tid) * GS + i) * GS + j] = v * rsqrtf(ss);
  }
}

// ------------------------------------------------------------------
extern "C" void kernel_launch(void* const* d_in, const int* in_sizes, int n_in,
                              void* d_out, int out_size, void* d_ws, size_t ws_size,
                              hipStream_t stream) {
  const float* x = (const float*)d_in[0];
  float* out = (float*)d_out;
  // output layout: clst | p2s | s2p
  float* clst = out;                                   // 2*32*16*16   = 16384
  float* p2s  = out + 16384;                           // 2*9*256*256  = 1179648
  float* s2p  = out + 16384 + 2 * 9 * HW;              // 2*2304*16*16 = 1179648
  // workspace: xn (16 MB) | pix_sim padded to 16 (8 MB)
  float* xn      = (float*)d_ws;
  float* pix_sim = xn + (size_t)B_ * C_ * HW;

  k_normalize<<<(B_ * HW) / 256, 256, 0, stream>>>(x, xn);
  k_init_clst<<<B_ * 256, 32, 0, stream>>>(x, clst);

  for (int it = 0; it < 5; ++it) {
    k_block_gemm<0><<<B_ * 256, 256, 0, stream>>>(xn, clst, pix_sim);
    k_update<<<B_ * 256, 256, 0, stream>>>(xn, pix_sim, clst, s2p);
  }
  k_block_gemm<1><<<B_ * 256, 256, 0, stream>>>(xn, clst, p2s);
}